// Edge2Node_40346922779128
// MI455X (gfx1250) — compile-verified
//
#include <hip/hip_runtime.h>

#define WAVES_PER_BLOCK 8
#define TILE_EDGES      8              // edges per wave-tile
#define ROW_BYTES       256            // F=64 floats * 4B
#define TILE_BYTES      (TILE_EDGES * ROW_BYTES)   // 2048
#define COPIES_PER_TILE (TILE_BYTES / 512)         // 4 x (32 lanes * 16B)
#define FDIM            64

// ---------------------------------------------------------------------------
// Zero-init of the output (segment_sum => empty segments are 0; atomics need
// a zeroed accumulator base; harness poisons d_out).
// ---------------------------------------------------------------------------
__global__ void zero_out_kernel(float* __restrict__ out, int n) {
    int i = blockIdx.x * blockDim.x + threadIdx.x;
    if (i < n) out[i] = 0.0f;
}

// LDS byte-offset of a __shared__ object (generic LDS pointers carry the
// offset in the low 32 bits per the gfx1250 aperture layout).
__device__ __forceinline__ unsigned lds_off(const void* p) {
    return (unsigned)(unsigned long long)p;
}

// CDNA5 async copy: global -> LDS, 16B per lane, tracked by ASYNCcnt.
// GVS mode: mem addr = SADDR(64b) + VADDR(32b offset); VDST holds LDS addr.
__device__ __forceinline__ void async_copy_b128(unsigned lds_addr,
                                                unsigned goff,
                                                const void* sbase) {
    asm volatile("global_load_async_to_lds_b128 %0, %1, %2 th:TH_LOAD_NT"
                 :: "v"(lds_addr), "v"(goff), "s"(sbase)
                 : "memory");
}

__device__ __forceinline__ void wait_async_le8() {
    asm volatile("s_wait_asynccnt 0x8" ::: "memory");
}
__device__ __forceinline__ void wait_async_0() {
    asm volatile("s_wait_asynccnt 0x0" ::: "memory");
}

// ---------------------------------------------------------------------------
// Scatter-add segment sum. Each wave double-buffers 2KB tiles of edge_w
// through LDS with async copies, then scatters float2 per lane via
// global_atomic_add_f32 (out is L2-resident: 12.8MB << 192MB L2; the NT hint
// on the 320MB edge_w stream keeps it from evicting out).
// ---------------------------------------------------------------------------
__global__ void edge2node_scatter_kernel(const int*   __restrict__ edge,
                                         const float* __restrict__ edge_w,
                                         const int*   __restrict__ dim_ptr,
                                         float*       __restrict__ out,
                                         int E, int num_tiles) {
    __shared__ __align__(16) unsigned char smem[WAVES_PER_BLOCK][2][TILE_BYTES];

    const int wave = threadIdx.x >> 5;
    const int lane = threadIdx.x & 31;

    const int row = (*dim_ptr == 1) ? 0 : 1;           // dim==1 -> group by edge[0]
    const int* __restrict__ seg_ids = edge + (size_t)row * (size_t)E;

    const unsigned long long gmax = (unsigned long long)E * (unsigned long long)ROW_BYTES;

    const int wave_stride = gridDim.x * WAVES_PER_BLOCK;
    const int t0 = blockIdx.x * WAVES_PER_BLOCK + wave;
    if (t0 >= num_tiles) return;                        // wave-uniform exit

    const unsigned base0 = lds_off(&smem[wave][0][0]);
    const unsigned base1 = lds_off(&smem[wave][1][0]);

    // issue one tile's async copies into the LDS buffer at lbase
    auto issue = [&](int tile, unsigned lbase) {
        unsigned gbase = (unsigned)tile * (unsigned)TILE_BYTES;
        #pragma unroll
        for (int it = 0; it < COPIES_PER_TILE; ++it) {
            unsigned off = gbase + (unsigned)it * 512u + (unsigned)lane * 16u;
            if ((unsigned long long)off + 16ull > gmax) off = 0;  // clamp tail reads
            async_copy_b128(lbase + (unsigned)it * 512u + (unsigned)lane * 16u,
                            off, edge_w);
        }
    };

    issue(t0, base0);                                   // prime the pipeline
    int buf = 0;

    for (int t = t0; t < num_tiles; t += wave_stride) {
        const int  tn        = t + wave_stride;
        const bool have_next = (tn < num_tiles);
        if (have_next) issue(tn, buf ? base0 : base1);  // overlap next HBM stream

        // segment ids for this tile (lanes replicate mod TILE_EDGES -> shfl source)
        const int eidx = t * TILE_EDGES + (lane & (TILE_EDGES - 1));
        const int seg  = (eidx < E) ? seg_ids[eidx] : -1;

        if (have_next) wait_async_le8();                // allows 8 pending (4+4): current not retired
        else           wait_async_0();

        const unsigned char* lp = buf ? &smem[wave][1][0] : &smem[wave][0][0];
        #pragma unroll
        for (int e = 0; e < TILE_EDGES; ++e) {
            const int s = __shfl(seg, e, 32);
            if (s < 0) continue;
            const float2 v = *(const float2*)(lp + e * ROW_BYTES + lane * 8);
            float* dst = out + (size_t)s * FDIM + lane * 2;  // 32 lanes = 256B row
            atomicAdd(dst + 0, v.x);
            atomicAdd(dst + 1, v.y);
        }
        buf ^= 1;
    }
    wait_async_0();
}

// ---------------------------------------------------------------------------
// Launch
// ---------------------------------------------------------------------------
extern "C" void kernel_launch(void* const* d_in, const int* in_sizes, int n_in,
                              void* d_out, int out_size, void* d_ws, size_t ws_size,
                              hipStream_t stream) {
    (void)d_ws; (void)ws_size; (void)n_in;

    const int*   edge    = (const int*)d_in[0];    // (2, E) int32
    const float* edge_w  = (const float*)d_in[1];  // (E, 64) float32
    const int*   dim_ptr = (const int*)d_in[5];    // python scalar -> 1-elem device array
    float*       out     = (float*)d_out;          // (N, 64) float32

    const int E = in_sizes[0] / 2;
    const int num_tiles = (E + TILE_EDGES - 1) / TILE_EDGES;

    // 1) zero the accumulator
    {
        const int threads = 256;
        const int blocks  = (out_size + threads - 1) / threads;
        zero_out_kernel<<<blocks, threads, 0, stream>>>(out, out_size);
    }

    // 2) async-pipelined scatter-add
    {
        const int threads = 32 * WAVES_PER_BLOCK;   // 256 = 8 waves
        int blocks = 1024;                          // ~8K wave streams, ~19 tiles each
        const int max_blocks = (num_tiles + WAVES_PER_BLOCK - 1) / WAVES_PER_BLOCK;
        if (blocks > max_blocks) blocks = max_blocks;
        if (blocks < 1) blocks = 1;
        edge2node_scatter_kernel<<<blocks, threads, 0, stream>>>(
            edge, edge_w, dim_ptr, out, E, num_tiles);
    }
}